// DsaScatterPatched_25666724561324
// MI455X (gfx1250) — compile-verified
//
#include <hip/hip_runtime.h>
#include <hip/hip_bf16.h>
#include <stdint.h>

// Problem constants (fixed by the reference): B=8, SQ=4096, SKV=4096, K=64.
#define SKV_C   4096
#define SQ_C    4096
#define KIDX    64
#define BLOCK   256
#define F4_ROW  (SKV_C / 4)        // 1024 float4 per row
#define ITERS   (F4_ROW / BLOCK)   // 4 x 16B per thread per row
#define LDS_BYTES (SKV_C * 4)      // 16 KB dynamic LDS row buffer @ offset 0

// One workgroup per (b, l) row. Row pipeline, entirely through CDNA5 async DMA:
//   global --global_load_async_to_lds_b128--> LDS
//         --ds_store scatter (asm)--> LDS
//         --global_store_async_from_lds_b128--> global
// 1 GiB total traffic => ~46us floor at 23.3 TB/s; everything else is noise.
// ALL LDS traffic is inline asm on raw byte offsets (dynamic LDS base = 0),
// so no C++-level LDS object exists for the optimizer to dead-store-eliminate.
__global__ __launch_bounds__(BLOCK) void dsa_scatter_row_kernel(
    const float* __restrict__ mask,   // (B, SQ, SKV)
    const int*   __restrict__ idx,    // (B, s1-s0, K)
    const int*   __restrict__ s0p,
    const int*   __restrict__ s1p,
    float*       __restrict__ out)    // (B, SQ, SKV)
{
    const int row = blockIdx.x;       // 0 .. B*SQ-1
    const int t   = threadIdx.x;
    const int s0  = s0p[0];
    const int s1  = s1p[0];
    const int b   = row / SQ_C;
    const int l   = row - b * SQ_C;

    const float* src = mask + (size_t)row * SKV_C;
    float*       dst = out  + (size_t)row * SKV_C;

    // ---- Stage-in: async DMA global -> LDS (B128 per lane, no VGPR data) ----
    #pragma unroll
    for (int i = 0; i < ITERS; ++i) {
        unsigned lds_off = (unsigned)((i * BLOCK + t) * 16);
        unsigned long long ga = (unsigned long long)(uintptr_t)src + lds_off;
        asm volatile("global_load_async_to_lds_b128 %0, %1, off"
                     :: "v"(lds_off), "v"(ga) : "memory");
    }
    asm volatile("s_wait_asynccnt 0" ::: "memory");  // our wave's DMA landed
    __syncthreads();                                 // everyone's DMA landed

    // ---- Scatter (wave 0 only: 2 indices per lane, wave32 ballots) ----
    const bool in_chunk = (l >= s0) && (l < s1);     // block-uniform
    if (in_chunk && t < 32) {
        const int chunk_len = s1 - s0;
        const int2* rowidx =
            (const int2*)(idx + ((size_t)b * chunk_len + (l - s0)) * KIDX);
        int2 v = rowidx[t];                          // lane covers k=2t, 2t+1

        const bool sent  = (v.x < 0)  || (v.y < 0);
        const bool real0 = (v.x == 0) || (v.y == 0);
        const bool any_sent  = __any((int)sent);     // wave32 ballot (full wave0 EXEC)
        const bool any_real0 = __any((int)real0);

        const int p0 = v.x < 0 ? 0 : v.x;
        const int p1 = v.y < 0 ? 0 : v.y;
        const float zero = 0.0f;
        if (p0 != 0) {                               // col 0 handled by lane 0 only
            unsigned a0 = (unsigned)(p0 * 4);
            asm volatile("ds_store_b32 %0, %1" :: "v"(a0), "v"(zero) : "memory");
        }
        if (p1 != 0) {
            unsigned a1 = (unsigned)(p1 * 4);
            asm volatile("ds_store_b32 %0, %1" :: "v"(a1), "v"(zero) : "memory");
        }
        if (t == 0 && (any_sent || any_real0)) {
            // real index 0 -> 0.0 ; sentinel only -> -inf (fixup) ; else untouched
            const float c0 = any_real0 ? 0.0f : -__builtin_inff();
            const unsigned az = 0u;
            asm volatile("ds_store_b32 %0, %1" :: "v"(az), "v"(c0) : "memory");
        }
    }
    asm volatile("s_wait_dscnt 0" ::: "memory");     // our asm ds stores complete
    __syncthreads();                                 // scatter visible to all waves

    // ---- Stage-out: async DMA LDS -> global ----
    #pragma unroll
    for (int i = 0; i < ITERS; ++i) {
        unsigned lds_off = (unsigned)((i * BLOCK + t) * 16);
        unsigned long long ga = (unsigned long long)(uintptr_t)dst + lds_off;
        asm volatile("global_store_async_from_lds_b128 %0, %1, off"
                     :: "v"(ga), "v"(lds_off) : "memory");
    }
    asm volatile("s_wait_asynccnt 0" ::: "memory");  // (s_endpgm would also wait)
}

extern "C" void kernel_launch(void* const* d_in, const int* in_sizes, int n_in,
                              void* d_out, int out_size, void* d_ws, size_t ws_size,
                              hipStream_t stream) {
    // setup_inputs order: index_mask, idx_chunk, finite_ref, finite_got, s0, s1
    const float* mask = (const float*)d_in[0];
    const int*   idx  = (const int*)d_in[1];
    const int*   s0p  = (const int*)d_in[4];
    const int*   s1p  = (const int*)d_in[5];
    float*       out  = (float*)d_out;

    const int rows = in_sizes[0] / SKV_C;  // B * SQ = 32768
    // 16 KB *dynamic* LDS: guarantees the allocation exists (offset 0) even
    // though no C++-level shared-memory object is referenced.
    dsa_scatter_row_kernel<<<rows, BLOCK, LDS_BYTES, stream>>>(
        mask, idx, s0p, s1p, out);
}